// Cross_Attention_36558761623717
// MI455X (gfx1250) — compile-verified
//
#include <hip/hip_runtime.h>
#include <math.h>

// MI455X / gfx1250: wave32, WMMA fp32 16x16x4, 320KB LDS per WGP.
// Memory-bound op (~1.1 GB traffic, ~27 GFLOP) -> fp32 WMMA chosen for accuracy.

#define BATCH 4
#define C     64
#define C3    192
#define HEADS 8
#define CHD   8
#define IMH   256
#define IMW   256
#define HW    65536

typedef __attribute__((ext_vector_type(2))) float v2f;
typedef __attribute__((ext_vector_type(8))) float v8f;

__device__ __forceinline__ v8f wmma_f32_k4(v2f a, v2f b, v8f c) {
  // D = A(16x4) * B(4x16) + C(16x16), fp32 everywhere.
  return __builtin_amdgcn_wmma_f32_16x16x4_f32(
      /*neg_a=*/false, a, /*neg_b=*/false, b,
      /*c_mod=*/(short)0, c, /*reuse_a=*/false, /*reuse_b=*/false);
}

// ---------------------------------------------------------------------------
// Kernel 1: fused conv1x1 (WMMA GEMM [192x64] x [64 x 18*18 halo]) -> LDS,
// then depthwise 3x3 from LDS, store qkv, accumulate q/k sum-of-squares.
// One block = one 16x16 pixel tile of one batch. LDS: 192*328*4 + 512 B.
// Zero-padding is done branchlessly (clamped address * mask) so the WMMA
// stream is never broken by EXEC save/restore sequences.
// ---------------------------------------------------------------------------
#define K1_STR 328   // padded LDS row stride (>= 324 halo pixels)

__global__ __launch_bounds__(256, 1)
void qkv_kernel(const float* __restrict__ in, const float* __restrict__ wqkv,
                const float* __restrict__ wdw, float* __restrict__ qkv,
                float* __restrict__ ssq) {
  extern __shared__ float smem[];
  float* conv = smem;                 // [192][K1_STR] conv1x1 output over halo
  float* ssql = smem + C3 * K1_STR;   // [128] local sum-of-squares (q,k)

  const int tid  = threadIdx.x;
  const int lane = tid & 31;
  const int wave = tid >> 5;
  const int b    = blockIdx.y;
  const int ty   = blockIdx.x >> 4;
  const int tx   = blockIdx.x & 15;
  const int h0   = ty * 16 - 1;       // halo origin (18x18 region)
  const int w0   = tx * 16 - 1;
  const float* inb = in + (size_t)b * C * HW;

  const int ln = lane & 15;
  const int kh = (lane < 16) ? 0 : 2; // K-pair selected by lane half

  // ---- Phase 1: conv1x1 via fp32 WMMA. 21 N-tiles (336 cols), 12 M-tiles.
  for (int nt = wave; nt < 21; nt += 8) {
    const int ncol = nt * 16 + ln;                 // halo pixel index (0..335)
    const int py = ncol / 18, px = ncol - py * 18; // halo coords
    const int gy = h0 + py, gx = w0 + px;
    const bool inr = (ncol < 324) && (gy >= 0) && (gy < IMH) &&
                     (gx >= 0) && (gx < IMW);
    const float maskf = inr ? 1.f : 0.f;           // branchless zero-pad
    const int gyc = min(max(gy, 0), IMH - 1);
    const int gxc = min(max(gx, 0), IMW - 1);
    const int gpixc = gyc * IMW + gxc;             // always-valid address

    // Hoist B fragments: 16 k-steps, reused across all 12 M-tiles.
    v2f bfrag[16];
#pragma unroll
    for (int i = 0; i < 16; ++i) {
      const float* p = inb + (size_t)(4 * i + kh) * HW + gpixc;
      bfrag[i].x = p[0]  * maskf;   // channel 4i+kh
      bfrag[i].y = p[HW] * maskf;   // channel 4i+kh+1
    }

    for (int mt = 0; mt < 12; ++mt) {
      const int arow = mt * 16 + ln;               // A-matrix row (M)
      v2f afrag[16];
#pragma unroll
      for (int i = 0; i < 16; ++i) {
        afrag[i].x = wqkv[arow * C + 4 * i + kh];
        afrag[i].y = wqkv[arow * C + 4 * i + kh + 1];
      }
      v8f acc = {};
#pragma unroll
      for (int i = 0; i < 16; ++i)                 // pure WMMA chain
        acc = wmma_f32_k4(afrag[i], bfrag[i], acc);

      // D (16x16): VGPR j -> rows mbase+j, col ncol
      const int mbase = mt * 16 + ((lane < 16) ? 0 : 8);
      if (ncol < 324) {
#pragma unroll
        for (int j = 0; j < 8; ++j)
          conv[(mbase + j) * K1_STR + ncol] = acc[j];
      }
    }
  }
  if (tid < 128) ssql[tid] = 0.f;
  __syncthreads();

  // ---- Phase 2: depthwise 3x3 from LDS halo, store, sum-of-squares for q/k.
  for (int idx = tid; idx < C3 * 256; idx += 256) {
    const int c = idx >> 8, p = idx & 255;
    const int oy = p >> 4, ox = p & 15;
    const float* base = conv + c * K1_STR + oy * 18 + ox;
    float s = 0.f;
#pragma unroll
    for (int dy = 0; dy < 3; ++dy)
#pragma unroll
      for (int dx = 0; dx < 3; ++dx)
        s = fmaf(wdw[c * 9 + dy * 3 + dx], base[dy * 18 + dx], s);
    const int gy = ty * 16 + oy, gx = tx * 16 + ox;
    qkv[((size_t)b * C3 + c) * HW + gy * IMW + gx] = s;
    if (c < 128) atomicAdd(&ssql[c], s * s);   // q: c<64, k: 64..127
  }
  __syncthreads();
  if (tid < 128) atomicAdd(&ssq[b * 128 + tid], ssql[tid]);
}

// ---------------------------------------------------------------------------
// Kernel 2: per-(b,head) 8x8 cross Gram over n=65536, normalize, softmax.
// pair 0: attn_a = softmax(+ (q_b . k_a)/(|q||k|) * temp)
// pair 1: attn_b = softmax(- (q_a . k_b)/(|q||k|) * temp)
// ---------------------------------------------------------------------------
__global__ __launch_bounds__(256, 1)
void attn_kernel(const float* __restrict__ qkvA, const float* __restrict__ qkvB,
                 const float* __restrict__ ssqA, const float* __restrict__ ssqB,
                 const float* __restrict__ temp,
                 float* __restrict__ attnA, float* __restrict__ attnB) {
  __shared__ float dred[CHD * CHD];
  const int tid = threadIdx.x;
  const int b = blockIdx.x >> 3, h = blockIdx.x & 7;

  for (int pair = 0; pair < 2; ++pair) {
    const float* q  = (pair ? qkvA : qkvB) + ((size_t)b * C3 + h * CHD) * HW;
    const float* k  = (pair ? qkvB : qkvA) + ((size_t)b * C3 + 64 + h * CHD) * HW;
    const float* sq = (pair ? ssqA : ssqB) + b * 128 + h * CHD;
    const float* sk = (pair ? ssqB : ssqA) + b * 128 + 64 + h * CHD;
    float* attn = pair ? attnB : attnA;
    const float sign = pair ? -1.f : 1.f;

    if (tid < CHD * CHD) dred[tid] = 0.f;
    __syncthreads();

    float acc[CHD][CHD];
#pragma unroll
    for (int c = 0; c < CHD; ++c)
#pragma unroll
      for (int d = 0; d < CHD; ++d) acc[c][d] = 0.f;

    for (int n = tid; n < HW; n += 256) {
      __builtin_prefetch(&q[(size_t)n + 2048], 0, 1);  // global_prefetch_b8
      __builtin_prefetch(&k[(size_t)n + 2048], 0, 1);
      float qv[CHD], kv[CHD];
#pragma unroll
      for (int c = 0; c < CHD; ++c) {
        qv[c] = q[(size_t)c * HW + n];
        kv[c] = k[(size_t)c * HW + n];
      }
#pragma unroll
      for (int c = 0; c < CHD; ++c)
#pragma unroll
        for (int d = 0; d < CHD; ++d)
          acc[c][d] = fmaf(qv[c], kv[d], acc[c][d]);
    }
#pragma unroll
    for (int c = 0; c < CHD; ++c)
#pragma unroll
      for (int d = 0; d < CHD; ++d)
        atomicAdd(&dred[c * CHD + d], acc[c][d]);
    __syncthreads();

    if (tid < CHD) {
      const int c = tid;
      const float qn = fmaxf(sqrtf(sq[c]), 1e-12f);
      const float tv = temp[h];
      float row[CHD];
      float mx = -3.4e38f;
#pragma unroll
      for (int d = 0; d < CHD; ++d) {
        const float kn = fmaxf(sqrtf(sk[d]), 1e-12f);
        row[d] = sign * (dred[c * CHD + d] / (qn * kn)) * tv;
        mx = fmaxf(mx, row[d]);
      }
      float se = 0.f;
#pragma unroll
      for (int d = 0; d < CHD; ++d) { row[d] = expf(row[d] - mx); se += row[d]; }
      const float inv = 1.f / se;
#pragma unroll
      for (int d = 0; d < CHD; ++d)
        attn[(((size_t)b * HEADS + h) * CHD + c) * CHD + d] = row[d] * inv;
    }
    __syncthreads();
  }
}

// ---------------------------------------------------------------------------
// Kernel 3: per 256-pixel chunk: t = attn @ v (8x8 block-diag, VALU) -> LDS,
// proj = Wproj(64x64) @ t + residual (WMMA) -> LDS,
// out  = Wcat(64x128) @ [proj_a; proj_b]   (WMMA) -> global.
// LDS: 4 * 64*260 floats + 1024 floats = 270 KB.
// ---------------------------------------------------------------------------
#define K3_STR 260

__global__ __launch_bounds__(256, 1)
void out_kernel(const float* __restrict__ x, const float* __restrict__ y,
                const float* __restrict__ qkvA, const float* __restrict__ qkvB,
                const float* __restrict__ attnA, const float* __restrict__ attnB,
                const float* __restrict__ wprojA, const float* __restrict__ wprojB,
                const float* __restrict__ wcat, float* __restrict__ out) {
  extern __shared__ float smem[];
  float* ta  = smem;
  float* tb  = ta + C * K3_STR;
  float* pa  = tb + C * K3_STR;
  float* pb  = pa + C * K3_STR;
  float* atA = pb + C * K3_STR;   // [512]
  float* atB = atA + 512;

  const int tid  = threadIdx.x;
  const int lane = tid & 31;
  const int wave = tid >> 5;
  const int b    = blockIdx.y;
  const int p0   = blockIdx.x * 256;

  for (int i = tid; i < 512; i += 256) {
    atA[i] = attnA[(size_t)b * 512 + i];
    atB[i] = attnB[(size_t)b * 512 + i];
  }
  __syncthreads();

  // ---- Phase A: t[c][p] = sum_d attn[h][cc][d] * v[h*8+d][p]
  for (int path = 0; path < 2; ++path) {
    const float* vsrc = (path ? qkvB : qkvA) + ((size_t)b * C3 + 128) * HW + p0;
    const float* at   = path ? atB : atA;
    float* tdst       = path ? tb : ta;
    for (int idx = tid; idx < HEADS * 256; idx += 256) {
      const int h = idx >> 8, p = idx & 255;
      float v[CHD];
#pragma unroll
      for (int d = 0; d < CHD; ++d)
        v[d] = vsrc[(size_t)(h * CHD + d) * HW + p];
#pragma unroll
      for (int cc = 0; cc < CHD; ++cc) {
        float s = 0.f;
#pragma unroll
        for (int d = 0; d < CHD; ++d)
          s = fmaf(at[(h * CHD + cc) * CHD + d], v[d], s);
        tdst[(h * CHD + cc) * K3_STR + p] = s;
      }
    }
  }
  __syncthreads();

  const int ln = lane & 15;
  const int kh = (lane < 16) ? 0 : 2;

  // ---- Phase B: proj (64x64) + residual, both paths. 128 (m,n) tile pairs.
  for (int tp = wave; tp < 128; tp += 8) {
    const int sel = tp >> 6, tt = tp & 63;
    const int mt = tt & 3, nt = tt >> 2;
    const float* Wp  = sel ? wprojB : wprojA;
    const float* res = (sel ? y : x) + (size_t)b * C * HW + p0;
    const float* src = sel ? tb : ta;
    float* dst       = sel ? pb : pa;
    const int nn   = nt * 16 + ln;
    const int arow = mt * 16 + ln;

    v2f afrag[16], bfrag[16];
#pragma unroll
    for (int i = 0; i < 16; ++i) {
      afrag[i].x = Wp[arow * C + 4 * i + kh];
      afrag[i].y = Wp[arow * C + 4 * i + kh + 1];
      bfrag[i].x = src[(4 * i + kh) * K3_STR + nn];
      bfrag[i].y = src[(4 * i + kh + 1) * K3_STR + nn];
    }
    v8f acc = {};
#pragma unroll
    for (int i = 0; i < 16; ++i)
      acc = wmma_f32_k4(afrag[i], bfrag[i], acc);

    const int mbase = mt * 16 + ((lane < 16) ? 0 : 8);
#pragma unroll
    for (int j = 0; j < 8; ++j)
      dst[(mbase + j) * K3_STR + nn] =
          acc[j] + res[(size_t)(mbase + j) * HW + nn];
  }
  __syncthreads();

  // ---- Phase C: concat conv (64 x 128), K split as [pa | pb] halves.
  for (int tp = wave; tp < 64; tp += 8) {
    const int mt = tp & 3, nt = tp >> 2;
    const int nn   = nt * 16 + ln;
    const int arow = mt * 16 + ln;
    v8f acc = {};
#pragma unroll
    for (int half = 0; half < 2; ++half) {
      const float* s = half ? pb : pa;
      v2f afrag[16], bfrag[16];
#pragma unroll
      for (int i = 0; i < 16; ++i) {
        const int k0 = half * C + 4 * i + kh;
        afrag[i].x = wcat[arow * 2 * C + k0];
        afrag[i].y = wcat[arow * 2 * C + k0 + 1];
        bfrag[i].x = s[(4 * i + kh) * K3_STR + nn];
        bfrag[i].y = s[(4 * i + kh + 1) * K3_STR + nn];
      }
#pragma unroll
      for (int i = 0; i < 16; ++i)
        acc = wmma_f32_k4(afrag[i], bfrag[i], acc);
    }
    const int mbase = mt * 16 + ((lane < 16) ? 0 : 8);
#pragma unroll
    for (int j = 0; j < 8; ++j)
      out[((size_t)b * C + mbase + j) * HW + p0 + nn] = acc[j];
  }
}

// ---------------------------------------------------------------------------
// Launch. Workspace layout (floats):
//   qkvA [4*192*65536] | qkvB [same] | ssqA[512] | ssqB[512]
//   attnA[2048] | attnB[2048]                          (~403 MB total)
// ---------------------------------------------------------------------------
extern "C" void kernel_launch(void* const* d_in, const int* in_sizes, int n_in,
                              void* d_out, int out_size, void* d_ws, size_t ws_size,
                              hipStream_t stream) {
  const float* x      = (const float*)d_in[0];
  const float* y      = (const float*)d_in[1];
  const float* temp   = (const float*)d_in[2];
  const float* wqkvA  = (const float*)d_in[3];
  const float* wdwA   = (const float*)d_in[4];
  const float* wqkvB  = (const float*)d_in[5];
  const float* wdwB   = (const float*)d_in[6];
  const float* wprojA = (const float*)d_in[7];
  const float* wprojB = (const float*)d_in[8];
  const float* wcat   = (const float*)d_in[9];
  float* out = (float*)d_out;

  float* ws    = (float*)d_ws;
  const size_t QKV = (size_t)BATCH * C3 * HW;   // 50,331,648 floats
  float* qkvA  = ws;
  float* qkvB  = ws + QKV;
  float* ssqA  = ws + 2 * QKV;
  float* ssqB  = ssqA + 512;
  float* attnA = ssqB + 512;
  float* attnB = attnA + 2048;

  hipMemsetAsync(ssqA, 0, 1024 * sizeof(float), stream);

  const dim3 g1(256, BATCH);
  const size_t lds1 = (size_t)(C3 * K1_STR + 128) * sizeof(float);  // ~252 KB
  qkv_kernel<<<g1, 256, lds1, stream>>>(x, wqkvA, wdwA, qkvA, ssqA);
  qkv_kernel<<<g1, 256, lds1, stream>>>(y, wqkvB, wdwB, qkvB, ssqB);

  attn_kernel<<<BATCH * HEADS, 256, 0, stream>>>(qkvA, qkvB, ssqA, ssqB,
                                                 temp, attnA, attnB);

  const dim3 g3(HW / 256, BATCH);
  const size_t lds3 = (size_t)(4 * C * K3_STR + 1024) * sizeof(float); // ~270 KB
  out_kernel<<<g3, 256, lds3, stream>>>(x, y, qkvA, qkvB, attnA, attnB,
                                        wprojA, wprojB, wcat, out);
}